// Shift_78666620993913
// MI455X (gfx1250) — compile-verified
//
#include <hip/hip_runtime.h>
#include <hip/hip_bf16.h>

// Shift-bilinear (stride==1) for (B,C,T,V) = (32,256,300,25) f32.
//
// Memory-bound: ~491 MB min traffic -> ~21us floor at 23.3 TB/s; ~0.6 GFLOP
// total, so no WMMA (no matmul structure). CDNA5 path used instead:
// per-(b,c) 30KB slice staged into LDS via GLOBAL_LOAD_ASYNC_TO_LDS_B128
// (ASYNCcnt), s_wait_asynccnt + workgroup barrier, 4-corner bilinear from
// LDS (per-channel-constant weights: frac(ypos[c]), frac(xpos[c])),
// non-temporal coalesced output stores.

#define T_DIM 300
#define V_DIM 25
#define SLICE (T_DIM * V_DIM)   // 7500 floats = 30000 B, 16B-aligned per slice
#define VEC4  (SLICE / 4)       // 1875 x b128 transfers
#define NTHREADS 256            // 8 wave32s

typedef float f4 __attribute__((ext_vector_type(4)));
typedef int   i4 __attribute__((ext_vector_type(4)));
typedef __attribute__((address_space(1))) i4 glb_i4;   // global (prints __device__)
typedef __attribute__((address_space(3))) i4 lds_i4;   // LDS    (prints __shared__)

__global__ __launch_bounds__(NTHREADS)
void shift_bilinear_f32(const float* __restrict__ in,
                        const float* __restrict__ xpos,
                        const float* __restrict__ ypos,
                        float* __restrict__ out,
                        int C)
{
    __shared__ __align__(16) float tile[SLICE];

    const int slice = blockIdx.x;               // slice = b*C + c
    const int c     = slice % C;
    const size_t base = (size_t)slice * SLICE;
    const float* __restrict__ gsrc = in  + base;
    float*       __restrict__ gdst = out + base;

    // ---- Stage the 300x25 slice into LDS -------------------------------
#if __has_builtin(__builtin_amdgcn_global_load_async_to_lds_b128)
    for (int j = threadIdx.x; j < VEC4; j += NTHREADS) {
        __builtin_amdgcn_global_load_async_to_lds_b128(
            (glb_i4*)(gsrc + 4 * j),
            (lds_i4*)(tile + 4 * j),
            /*imm offset*/ 0, /*cpol*/ 0);
    }
#if __has_builtin(__builtin_amdgcn_s_wait_asynccnt)
    __builtin_amdgcn_s_wait_asynccnt(0);
#else
    asm volatile("s_wait_asynccnt 0" ::: "memory");
#endif
    __syncthreads();
#else
    // Fallback: global_load_b128 -> ds_store_b128 (still coalesced staging).
    for (int j = threadIdx.x; j < VEC4; j += NTHREADS) {
        const f4 d = __builtin_nontemporal_load(((const f4*)gsrc) + j);
        ((f4*)tile)[j] = d;
    }
    __syncthreads();
#endif

    // ---- Per-channel-constant bilinear weights -------------------------
    // t,v are integers => floor(t+ypos) = t + floor(ypos), wy = frac(ypos).
    const float yp = ypos[c];
    const float xp = xpos[c];
    const float fy = floorf(yp);
    const float fx = floorf(xp);
    const int   iy = (int)fy;
    const int   ix = (int)fx;
    const float wy = yp - fy;
    const float wx = xp - fx;
    const float w00 = (1.0f - wy) * (1.0f - wx);
    const float w01 = (1.0f - wy) * wx;
    const float w10 = wy * (1.0f - wx);
    const float w11 = wy * wx;

    // ---- 4-corner gather from LDS, coalesced NT store ------------------
    for (int idx = threadIdx.x; idx < SLICE; idx += NTHREADS) {
        const int t = idx / V_DIM;
        const int v = idx - t * V_DIM;

        const int y0 = t + iy, y1 = y0 + 1;
        const int x0 = v + ix, x1 = x0 + 1;

        const bool vy0 = (unsigned)y0 < (unsigned)T_DIM;
        const bool vy1 = (unsigned)y1 < (unsigned)T_DIM;
        const bool vx0 = (unsigned)x0 < (unsigned)V_DIM;
        const bool vx1 = (unsigned)x1 < (unsigned)V_DIM;

        const int r0 = (vy0 ? y0 : 0) * V_DIM;
        const int r1 = (vy1 ? y1 : 0) * V_DIM;
        const int c0 = vx0 ? x0 : 0;
        const int c1 = vx1 ? x1 : 0;

        const float v00 = (vy0 && vx0) ? tile[r0 + c0] : 0.0f;
        const float v01 = (vy0 && vx1) ? tile[r0 + c1] : 0.0f;
        const float v10 = (vy1 && vx0) ? tile[r1 + c0] : 0.0f;
        const float v11 = (vy1 && vx1) ? tile[r1 + c1] : 0.0f;

        float r = w00 * v00;
        r = fmaf(w01, v01, r);
        r = fmaf(w10, v10, r);
        r = fmaf(w11, v11, r);

        __builtin_nontemporal_store(r, gdst + idx);
    }
}

extern "C" void kernel_launch(void* const* d_in, const int* in_sizes, int n_in,
                              void* d_out, int out_size, void* d_ws, size_t ws_size,
                              hipStream_t stream) {
    const float* in   = (const float*)d_in[0];
    const float* xpos = (const float*)d_in[1];
    const float* ypos = (const float*)d_in[2];
    // d_in[3] is `stride` (==1 for this setup); the stride==1 math is baked in.
    float* out = (float*)d_out;

    const int C       = in_sizes[1];            // 256
    const int nslices = in_sizes[0] / SLICE;    // B*C = 8192

    shift_bilinear_f32<<<nslices, NTHREADS, 0, stream>>>(in, xpos, ypos, out, C);
}